// NeuralTangentKernel_15006615733870
// MI455X (gfx1250) — compile-verified
//
#include <hip/hip_runtime.h>

// NTK closed form:
//   K[i,j] = (x1_i . x2_j)*(g1_i . g2_j) + (g1_i . g2_j) + (h1_i . h2_j) + 1
// with h = relu(x@W1 + b1), g = (x@W1 + b1 > 0) * W2[:,0].
// All GEMMs done with full-precision V_WMMA_F32_16X16X4_F32 (wave32, one
// 16x16 output tile per wave).

typedef __attribute__((ext_vector_type(2))) float v2f;
typedef __attribute__((ext_vector_type(8))) float v8f;

#define NS    256
#define D_IN  512
#define D_HID 1024

__device__ __forceinline__ v8f wmma_f32(v2f a, v2f b, v8f c) {
    // 8 args: (neg_a, A, neg_b, B, c_mod, C, reuse_a, reuse_b)
    return __builtin_amdgcn_wmma_f32_16x16x4_f32(
        false, a, false, b, (short)0, c, false, false);
}

// ---------------------------------------------------------------------------
// Kernel A: Z = X @ W1 + b1  ->  H = relu(Z), G = (Z>0)*W2col
// Grid (D_HID/16, NS/16), block = 32 (one wave per 16x16 tile).
//
// A-matrix (16x4 f32, 2 VGPRs/lane): lane<16 holds X[m0+lane][k..k+1],
// lane>=16 holds X[m0+lane-16][k+2..k+3]  -> contiguous float2 load.
// B-matrix (4x16 f32): lane<16 holds W1[k..k+1][n0+lane] in v0,v1;
// lane>=16 holds W1[k+2..k+3][n0+lane-16] -> two column-strided loads.
// C/D layout: VGPR r, lane l -> row m0 + r + 8*(l>>4), col n0 + (l&15).
// ---------------------------------------------------------------------------
__global__ void __launch_bounds__(32)
ntk_feat_kernel(const float* __restrict__ X,  const float* __restrict__ W1,
                const float* __restrict__ b1, const float* __restrict__ W2,
                float* __restrict__ H, float* __restrict__ G)
{
    const int lane = threadIdx.x;
    const int half = lane >> 4;     // 0: lanes 0-15, 1: lanes 16-31
    const int lm   = lane & 15;
    const int m0 = blockIdx.y * 16;
    const int n0 = blockIdx.x * 16;

    const float* arow = X  + (size_t)(m0 + lm) * D_IN + 2 * half;
    const float* bcol = W1 + (size_t)(2 * half) * D_HID + (n0 + lm);

    v8f c = {};
#pragma unroll 4
    for (int k = 0; k < D_IN; k += 4) {
        v2f a = *(const v2f*)(arow + k);
        v2f b;
        b.x = bcol[(size_t)(k)     * D_HID];
        b.y = bcol[(size_t)(k + 1) * D_HID];
        c = wmma_f32(a, b, c);
    }

    const int   col  = n0 + lm;
    const float bias = b1[col];
    const float w2c  = W2[col];
#pragma unroll
    for (int r = 0; r < 8; ++r) {
        const int row = m0 + r + 8 * half;
        const float z = c[r] + bias;
        H[(size_t)row * D_HID + col] = (z > 0.0f) ? z   : 0.0f;
        G[(size_t)row * D_HID + col] = (z > 0.0f) ? w2c : 0.0f;
    }
}

// ---------------------------------------------------------------------------
// Kernel B: fused Grams + combine.
//   Sxx = X1 @ X2^T (K=512), Sgg = G1 @ G2^T, Shh = H1 @ H2^T (K=1024)
//   out = Sxx*Sgg + Sgg + Shh + 1
// For B = Mat2^T, B[k][n] = Mat2[j0+n][k], so the B fragment is also a
// contiguous float2 load of Mat2's row (lane indexes N, VGPR pair indexes K).
// Grid (NS/16, NS/16), block = 32.
// ---------------------------------------------------------------------------
__global__ void __launch_bounds__(32)
ntk_gram_kernel(const float* __restrict__ X1, const float* __restrict__ X2,
                const float* __restrict__ H1, const float* __restrict__ H2,
                const float* __restrict__ G1, const float* __restrict__ G2,
                float* __restrict__ out)
{
    const int lane = threadIdx.x;
    const int half = lane >> 4;
    const int lm   = lane & 15;
    const int i0 = blockIdx.y * 16;   // rows: x1 samples
    const int j0 = blockIdx.x * 16;   // cols: x2 samples

    // --- Sxx over K = D_IN ---
    const float* a_x = X1 + (size_t)(i0 + lm) * D_IN + 2 * half;
    const float* b_x = X2 + (size_t)(j0 + lm) * D_IN + 2 * half;
    v8f cxx = {};
#pragma unroll 4
    for (int k = 0; k < D_IN; k += 4) {
        v2f a = *(const v2f*)(a_x + k);
        v2f b = *(const v2f*)(b_x + k);
        cxx = wmma_f32(a, b, cxx);
    }

    // --- Sgg, Shh over K = D_HID (shared loop) ---
    const float* a_h = H1 + (size_t)(i0 + lm) * D_HID + 2 * half;
    const float* b_h = H2 + (size_t)(j0 + lm) * D_HID + 2 * half;
    const float* a_g = G1 + (size_t)(i0 + lm) * D_HID + 2 * half;
    const float* b_g = G2 + (size_t)(j0 + lm) * D_HID + 2 * half;
    v8f chh = {}, cgg = {};
#pragma unroll 2
    for (int k = 0; k < D_HID; k += 4) {
        v2f ah = *(const v2f*)(a_h + k);
        v2f bh = *(const v2f*)(b_h + k);
        chh = wmma_f32(ah, bh, chh);
        v2f ag = *(const v2f*)(a_g + k);
        v2f bg = *(const v2f*)(b_g + k);
        cgg = wmma_f32(ag, bg, cgg);
    }

    // --- combine + store: K = Sxx*Sgg + Sgg + Shh + 1 ---
#pragma unroll
    for (int r = 0; r < 8; ++r) {
        const int row = i0 + r + 8 * half;
        const int col = j0 + lm;
        out[(size_t)row * NS + col] =
            cxx[r] * cgg[r] + cgg[r] + chh[r] + 1.0f;
    }
}

extern "C" void kernel_launch(void* const* d_in, const int* in_sizes, int n_in,
                              void* d_out, int out_size, void* d_ws, size_t ws_size,
                              hipStream_t stream) {
    const float* x1 = (const float*)d_in[0];   // [256, 512]
    const float* x2 = (const float*)d_in[1];   // [256, 512]
    const float* W1 = (const float*)d_in[2];   // [512, 1024]
    const float* b1 = (const float*)d_in[3];   // [1024]
    const float* W2 = (const float*)d_in[4];   // [1024, 1] -> treated as [1024]
    // d_in[5] = b2: does not affect the NTK (d out / d b2 == 1 always).

    float* H1 = (float*)d_ws;                  // 4 x 1MB scratch
    float* H2 = H1 + (size_t)NS * D_HID;
    float* G1 = H2 + (size_t)NS * D_HID;
    float* G2 = G1 + (size_t)NS * D_HID;

    const dim3 fgrid(D_HID / 16, NS / 16);     // (64, 16)
    ntk_feat_kernel<<<fgrid, 32, 0, stream>>>(x1, W1, b1, W2, H1, G1);
    ntk_feat_kernel<<<fgrid, 32, 0, stream>>>(x2, W1, b1, W2, H2, G2);

    const dim3 ggrid(NS / 16, NS / 16);        // (16, 16)
    ntk_gram_kernel<<<ggrid, 32, 0, stream>>>(x1, x2, H1, H2, G1, G2,
                                              (float*)d_out);
}